// Spiking3DPoolingLayer_11381663334469
// MI455X (gfx1250) — compile-verified
//
#include <hip/hip_runtime.h>
#include <hip/hip_bf16.h>

typedef __attribute__((ext_vector_type(2))) float v2f;
typedef __attribute__((ext_vector_type(8))) float v8f;

// ---------------------------------------------------------------------------
// Pass 1: weighted temporal reduction  S[p] = sum_t (50-t)*x[t][p]  via
// V_WMMA_F32_16X16X4_F32, then per-2x2-window argmax -> flat H*W index.
//
// One block per (b,c) plane (512 blocks, 256 threads = 8 waves).
// Per wave-assignment: one window-row segment = rows {2ho, 2ho+1},
// 16 consecutive columns. Two f32 accumulators (top row / bottom row).
// A-matrix per chunk: 16 pixels (M) x 4 timesteps (K); documented layout:
//   lanes 0-15 : M = lane,  v0 -> K=0 (t0+0), v1 -> K=1 (t0+1)
//   lanes 16-31: M = lane-16, v0 -> K=2 (t0+2), v1 -> K=3 (t0+3)
// B-matrix: B[k,n] = w[t0+k] for all n (K mapping mirrors A), so
// D[m,n] = weighted sum of pixel m, replicated over n.
// D layout: VGPR r holds M=r (lanes 0-15) / M=8+r (lanes 16-31), so lane 0
// holds sums for cols col0..col0+7 and lane 16 for col0+8..col0+15:
// each resolves 4 windows locally.
// ---------------------------------------------------------------------------
__global__ void __launch_bounds__(256)
spk_wsum_argmax_kernel(const float* __restrict__ x, int* __restrict__ idx_out) {
    const int T = 50, HW = 64 * 64, W = 64;
    const int bc   = blockIdx.x;             // 0..511  (b*64+c)
    const int lane = threadIdx.x & 31;
    const int wave = threadIdx.x >> 5;       // 0..7
    const int half = lane >> 4;              // 0: K={0,1}, 1: K={2,3}
    const int lm   = lane & 15;              // pixel (M) within group

    const float* plane = x + (size_t)bc * T * HW;

    // 128 assignments per plane: 32 window-rows x 4 column segments of 16
    for (int a = wave; a < 128; a += 8) {
        const int ho   = a >> 2;             // window row 0..31
        const int cs   = a & 3;              // column segment 0..3
        const int col0 = cs * 16;
        const int rowT = 2 * ho;
        const int offT = rowT * W + col0 + lm;       // top-row pixel offset
        const int offB = offT + W;                   // bottom-row pixel offset

        v8f ct = {};   // weighted sums, top row
        v8f cb = {};   // weighted sums, bottom row

        for (int t0 = 0; t0 < T; t0 += 4) {          // 13 chunks (tail padded)
            const int ta = t0 + 2 * half;            // K slot in v0
            const int tb = ta + 1;                   // K slot in v1
            v2f aT, aB, bw;
            aT.x = (ta < T) ? plane[(size_t)ta * HW + offT] : 0.0f;
            aT.y = (tb < T) ? plane[(size_t)tb * HW + offT] : 0.0f;
            aB.x = (ta < T) ? plane[(size_t)ta * HW + offB] : 0.0f;
            aB.y = (tb < T) ? plane[(size_t)tb * HW + offB] : 0.0f;
            bw.x = (ta < T) ? (float)(T - ta) : 0.0f;
            bw.y = (tb < T) ? (float)(T - tb) : 0.0f;
            ct = __builtin_amdgcn_wmma_f32_16x16x4_f32(false, aT, false, bw,
                                                       (short)0, ct, false, false);
            cb = __builtin_amdgcn_wmma_f32_16x16x4_f32(false, aB, false, bw,
                                                       (short)0, cb, false, false);
        }

        // Lane 0 owns cols [col0, col0+8); lane 16 owns [col0+8, col0+16).
        if (lm == 0) {
            const int cbase = col0 + 8 * half;
            for (int wl = 0; wl < 4; ++wl) {
                const float v0 = ct[2 * wl];         // (dh,dw)=(0,0)
                const float v1 = ct[2 * wl + 1];     // (0,1)
                const float v2 = cb[2 * wl];         // (1,0)
                const float v3 = cb[2 * wl + 1];     // (1,1)
                int e = 0; float best = v0;          // first-max (torch)
                if (v1 > best) { best = v1; e = 1; }
                if (v2 > best) { best = v2; e = 2; }
                if (v3 > best) { best = v3; e = 3; }
                const int dh = e >> 1, dw = e & 1;
                const int flat = (rowT + dh) * W + (cbase + 2 * wl + dw);
                const int wo   = (cbase >> 1) + wl;  // global window col 0..31
                idx_out[bc * 1024 + ho * 32 + wo] = flat;
            }
        }
    }
}

// ---------------------------------------------------------------------------
// Pass 2: gather winning pixel at every timestep.
// One thread per (b,c,ho,wo); loops t (index reused 50x). Writes coalesced.
// ---------------------------------------------------------------------------
__global__ void __launch_bounds__(256)
spk_gather_kernel(const float* __restrict__ x, const int* __restrict__ idx,
                  float* __restrict__ out, float* __restrict__ loss) {
    const int tid  = blockIdx.x * blockDim.x + threadIdx.x;  // 0..524287
    const int bc   = tid >> 10;
    const int howo = tid & 1023;
    const int f    = idx[tid];

    const float* px = x   + (size_t)bc * 50 * 4096;
    float*       po = out + (size_t)bc * 50 * 1024;
#pragma unroll 5
    for (int t = 0; t < 50; ++t) {
        po[(size_t)t * 1024 + howo] = px[(size_t)t * 4096 + f];
    }
    if (tid == 0) *loss = 0.0f;                // second tuple output
}

extern "C" void kernel_launch(void* const* d_in, const int* in_sizes, int n_in,
                              void* d_out, int out_size, void* d_ws, size_t ws_size,
                              hipStream_t stream) {
    (void)in_sizes; (void)n_in; (void)ws_size;
    const float* x    = (const float*)d_in[0];            // (8,64,50,64,64) f32
    float*       out  = (float*)d_out;                    // spk_rec flat + loss
    float*       loss = out + (out_size - 1);
    int*         idxb = (int*)d_ws;                       // 512*1024 ints = 2 MB

    // Pass 1: weighted sum + argmax indices (reads x once: ~419 MB)
    spk_wsum_argmax_kernel<<<512, 256, 0, stream>>>(x, idxb);
    // Pass 2: gather per timestep (~105 MB read + ~105 MB write)
    spk_gather_kernel<<<2048, 256, 0, stream>>>(x, idxb, out, loss);
}